// MetaBaseClassifier_31559419691703
// MI455X (gfx1250) — compile-verified
//
#include <hip/hip_runtime.h>
#include <hip/hip_bf16.h>
#include <math.h>

typedef __attribute__((ext_vector_type(2))) float v2f;
typedef __attribute__((ext_vector_type(4))) float v4f;
typedef __attribute__((ext_vector_type(8))) float v8f;

#define NB    512
#define NC    2048
#define NHW   48
#define NHID  128
#define NATTR 51
#define BNEPS 1e-5f

// ---------------------------------------------------------------------------
// Kernel 1: fused streaming pass over body/face (402 MB -> memory bound).
//   pooled[b,c] = max_hw(body * A_front * face)
//   meanb [b,c] = mean_hw(body)
// One thread per (b,c) row: 12 x float4 NT loads from each big tensor.
// ---------------------------------------------------------------------------
__global__ __launch_bounds__(256) void k_fuse_pool(
    const float* __restrict__ body, const float* __restrict__ face,
    const float* __restrict__ Af, float* __restrict__ pooled,
    float* __restrict__ meanb)
{
    int row = blockIdx.x * 256 + threadIdx.x;   // row = b*C + c
    int c   = row & (NC - 1);
    const v4f* pb = (const v4f*)(body + (size_t)row * NHW);
    const v4f* pf = (const v4f*)(face + (size_t)row * NHW);
    const v4f* pa = (const v4f*)(Af   + (size_t)c   * NHW);

    float mx = -3.402823466e38f;
    float sm = 0.f;
#pragma unroll
    for (int i = 0; i < NHW / 4; ++i) {
        v4f vb = __builtin_nontemporal_load(pb + i);   // streaming: TH_NT
        v4f vf = __builtin_nontemporal_load(pf + i);
        v4f va = pa[i];                                // L2-resident (384 KB)
        v4f p  = vb * va * vf;
        mx = fmaxf(mx, fmaxf(fmaxf(p.x, p.y), fmaxf(p.z, p.w)));
        sm += (vb.x + vb.y) + (vb.z + vb.w);
    }
    pooled[row] = mx;
    meanb[row]  = sm * (1.0f / NHW);
}

// ---------------------------------------------------------------------------
// WMMA f32 16x16x4 layouts (wave32):
//   A 16x4 : lane L, vgpr v -> M = L%16, K = 2*(L/16) + v   (contiguous K pair)
//   B 4x16 : lane L, vgpr v -> N = L%16, K = 2*(L/16) + v
//   C/D    : vgpr r, lanes 0-15: (M=r, N=lane); lanes 16-31: (M=r+8, N=lane-16)
// ---------------------------------------------------------------------------

// Kernel 2: h = relu(pooled @ W1^T + b1)   [512,2048]x[2048,128] -> [512,128]
// 8 waves/block, 1 tile/wave; 32 blocks = 256 tiles (32 Mt x 8 Nt).
__global__ __launch_bounds__(256) void k_gemm_gate1(
    const float* __restrict__ pooled, const float* __restrict__ W1,
    const float* __restrict__ b1, float* __restrict__ h)
{
    int lane = threadIdx.x & 31, half = lane >> 4, lm = lane & 15;
    int tile = blockIdx.x * 8 + (threadIdx.x >> 5);
    int tm = tile >> 3, tn = tile & 7;

    const float* pA = pooled + (size_t)(tm * 16 + lm) * NC + 2 * half;
    const float* pB = W1     + (size_t)(tn * 16 + lm) * NC + 2 * half;

    v8f acc = {};
#pragma unroll 8
    for (int k = 0; k < NC; k += 4) {
        v2f a = *(const v2f*)(pA + k);
        v2f b = *(const v2f*)(pB + k);
        acc = __builtin_amdgcn_wmma_f32_16x16x4_f32(
            false, a, false, b, (short)0, acc, false, false);
    }
    int col = tn * 16 + lm;
    float bias = b1[col];
#pragma unroll
    for (int r = 0; r < 8; ++r) {
        int m = tm * 16 + r + 8 * half;
        float v = acc[r] + bias;
        h[(size_t)m * NHID + col] = v > 0.f ? v : 0.f;
    }
}

// Kernel 3: gate = sigmoid(h @ W2^T + b2);  gap = meanb + (pose==1 ? gate : 0)
// [512,128]x[128,2048] -> [512,2048].  4096 tiles (32 Mt x 128 Nt), 512 blocks.
__global__ __launch_bounds__(256) void k_gemm_gate2(
    const float* __restrict__ h, const float* __restrict__ W2,
    const float* __restrict__ b2, const int* __restrict__ pose,
    const float* __restrict__ meanb, float* __restrict__ gap)
{
    int lane = threadIdx.x & 31, half = lane >> 4, lm = lane & 15;
    int tile = blockIdx.x * 8 + (threadIdx.x >> 5);
    int tm = tile >> 7, tn = tile & 127;

    const float* pA = h  + (size_t)(tm * 16 + lm) * NHID + 2 * half;
    const float* pB = W2 + (size_t)(tn * 16 + lm) * NHID + 2 * half;

    v8f acc = {};
#pragma unroll
    for (int k = 0; k < NHID; k += 4) {
        v2f a = *(const v2f*)(pA + k);
        v2f b = *(const v2f*)(pB + k);
        acc = __builtin_amdgcn_wmma_f32_16x16x4_f32(
            false, a, false, b, (short)0, acc, false, false);
    }
    int col = tn * 16 + lm;
    float bias = b2[col];
#pragma unroll
    for (int r = 0; r < 8; ++r) {
        int brow = tm * 16 + r + 8 * half;
        float g  = 1.f / (1.f + __expf(-(acc[r] + bias)));
        float add = (pose[brow] == 1) ? g : 0.f;
        size_t idx = (size_t)brow * NC + col;
        gap[idx] = meanb[idx] + add;
    }
}

// Kernel 4: z = gap @ Wl^T + bl   [512,2048]x[2048,51] -> [512,51]
// N padded to 64 (4 Nt); branchless zero-mask on B keeps EXEC all-ones.
// 128 tiles (32 Mt x 4 Nt), 16 blocks.
__global__ __launch_bounds__(256) void k_gemm_logits(
    const float* __restrict__ gap, const float* __restrict__ Wl,
    const float* __restrict__ bl, float* __restrict__ z)
{
    int lane = threadIdx.x & 31, half = lane >> 4, lm = lane & 15;
    int tile = blockIdx.x * 8 + (threadIdx.x >> 5);
    int tm = tile >> 2, tn = tile & 3;

    int n = tn * 16 + lm;
    int nc = n < NATTR ? n : 0;
    float msk = n < NATTR ? 1.f : 0.f;

    const float* pA = gap + (size_t)(tm * 16 + lm) * NC + 2 * half;
    const float* pB = Wl  + (size_t)nc * NC + 2 * half;

    v8f acc = {};
#pragma unroll 8
    for (int k = 0; k < NC; k += 4) {
        v2f a = *(const v2f*)(pA + k);
        v2f b = *(const v2f*)(pB + k);
        b.x *= msk; b.y *= msk;
        acc = __builtin_amdgcn_wmma_f32_16x16x4_f32(
            false, a, false, b, (short)0, acc, false, false);
    }
    if (n < NATTR) {
        float bias = bl[n];
#pragma unroll
        for (int r = 0; r < 8; ++r) {
            int brow = tm * 16 + r + 8 * half;
            z[(size_t)brow * NATTR + n] = acc[r] + bias;
        }
    }
}

// Kernel 5: BatchNorm1d (training mode, batch stats) over dim 0 of z[512,51].
__global__ __launch_bounds__(256) void k_bnorm(
    const float* __restrict__ z, const float* __restrict__ gamma,
    const float* __restrict__ beta, float* __restrict__ out)
{
    __shared__ float s1[256], s2[256];
    int j = blockIdx.x, t = threadIdx.x;
    float a = z[(size_t)t * NATTR + j];
    float b = z[(size_t)(t + 256) * NATTR + j];
    s1[t] = a + b;
    s2[t] = a * a + b * b;
    __syncthreads();
#pragma unroll
    for (int off = 128; off > 0; off >>= 1) {
        if (t < off) { s1[t] += s1[t + off]; s2[t] += s2[t + off]; }
        __syncthreads();
    }
    float mu  = s1[0] * (1.f / NB);
    float var = s2[0] * (1.f / NB) - mu * mu;
    float sc  = gamma[j] * rsqrtf(var + BNEPS);
    float bt  = beta[j];
    out[(size_t)t * NATTR + j]         = (a - mu) * sc + bt;
    out[(size_t)(t + 256) * NATTR + j] = (b - mu) * sc + bt;
}

// ---------------------------------------------------------------------------
extern "C" void kernel_launch(void* const* d_in, const int* in_sizes, int n_in,
                              void* d_out, int out_size, void* d_ws, size_t ws_size,
                              hipStream_t stream)
{
    const float* body  = (const float*)d_in[0];
    const float* face  = (const float*)d_in[1];
    const int*   pose  = (const int*)  d_in[2];
    const float* Af    = (const float*)d_in[3];
    const float* W1    = (const float*)d_in[4];
    const float* b1    = (const float*)d_in[5];
    const float* W2    = (const float*)d_in[6];
    const float* b2    = (const float*)d_in[7];
    const float* Wl    = (const float*)d_in[8];
    const float* bl    = (const float*)d_in[9];
    const float* gamma = (const float*)d_in[10];
    const float* beta  = (const float*)d_in[11];
    float* out = (float*)d_out;

    // workspace layout (floats): ~12.9 MB total
    float* ws     = (float*)d_ws;
    float* pooled = ws;                              // B*C = 1,048,576
    float* meanb  = pooled + (size_t)NB * NC;        // B*C
    float* h      = meanb  + (size_t)NB * NC;        // B*HID = 65,536
    float* gap    = h      + (size_t)NB * NHID;      // B*C
    float* z      = gap    + (size_t)NB * NC;        // B*NATTR = 26,112

    k_fuse_pool  <<<(NB * NC) / 256, 256, 0, stream>>>(body, face, Af, pooled, meanb);
    k_gemm_gate1 <<<32, 256, 0, stream>>>(pooled, W1, b1, h);
    k_gemm_gate2 <<<512, 256, 0, stream>>>(h, W2, b2, pose, meanb, gap);
    k_gemm_logits<<<16, 256, 0, stream>>>(gap, Wl, bl, z);
    k_bnorm      <<<NATTR, 256, 0, stream>>>(z, gamma, beta, out);
}